// STARLoss_23914377904685
// MI455X (gfx1250) — compile-verified
//
#include <hip/hip_runtime.h>
#include <hip/hip_bf16.h>
#include <math.h>

typedef float v2f __attribute__((ext_vector_type(2)));
typedef float v8f __attribute__((ext_vector_type(8)));

#define EPSL 1e-5f

__device__ __forceinline__ float waveReduceAdd(float v) {
#pragma unroll
    for (int off = 16; off >= 1; off >>= 1)
        v += __shfl_xor(v, off, 32);
    return v;
}

// Stage 1: one wave32 per (b,n) pair. WMMA f32 16x16x4 used as a weighted
// column-reduction engine over the 64x64 heatmap.
__global__ __launch_bounds__(256) void star_stage1(
    const float* __restrict__ hm, const float* __restrict__ gt,
    float* __restrict__ ws, int n_pairs)
{
    const int lane = threadIdx.x & 31;
    const int wave = threadIdx.x >> 5;
    const int p    = blockIdx.x * 8 + wave;

    __shared__ float sT[8];
    __shared__ float sE[8];

    float tcontrib = 0.0f, econtrib = 0.0f;

    if (p < n_pairs) {
        const float* __restrict__ map = hm + (size_t)p * 4096;
        const int n   = lane & 15;   // A's M row / B's N column / D's N column
        const int hi  = lane >> 4;   // selects K rows {0,1} vs {2,3} of the 4-slab
        const float stepc = 2.0f / 63.0f;

        float s0 = 0.f, sx = 0.f, sxx = 0.f, sy = 0.f, sxy = 0.f, syy = 0.f, s2 = 0.f;

        // Precompute A-weight selection masks (lane-invariant across loop)
        const bool m0 = (n == 0), m1 = (n == 1), m2 = (n == 2);

        for (int cb = 0; cb < 4; ++cb) {
            v8f c = {0.f, 0.f, 0.f, 0.f, 0.f, 0.f, 0.f, 0.f};
            const int col = cb * 16 + n;
            const float* __restrict__ mp = map + col;

#pragma unroll 4
            for (int kb = 0; kb < 16; ++kb) {
                const int ra = kb * 4 + 2 * hi;      // absolute heatmap rows ra, ra+1
                // B fragment: 4x16 heatmap tile, coalesced across lanes 0-15 / 16-31
                float bx = __builtin_nontemporal_load(mp + ra * 64);
                float by = __builtin_nontemporal_load(mp + (ra + 1) * 64);
                s2 = fmaf(bx, bx, s2);
                s2 = fmaf(by, by, s2);

                // A fragment: rows 0,1,2 = {1, y, y^2}; rows 3..15 = 0
                float ya = (float)ra * stepc - 1.0f;
                float yb = ya + stepc;
                float ax = m0 ? 1.0f : (m1 ? ya : (m2 ? ya * ya : 0.0f));
                float ay = m0 ? 1.0f : (m1 ? yb : (m2 ? yb * yb : 0.0f));

                v2f a;  a.x  = ax; a.y  = ay;
                v2f bv; bv.x = bx; bv.y = by;
                // D(16x16,f32) = A(16x4,f32) * B(4x16,f32) + C
                c = __builtin_amdgcn_wmma_f32_16x16x4_f32(
                        false, a, false, bv, (short)0, c, false, false);
            }
            // Lanes 0-15: c[v] = D[v][n]  (v=0: colsum, v=1: y-weighted, v=2: y^2-weighted)
            // Lanes 16-31 hold D rows 8-15 which are exactly zero.
            float x    = (float)col * stepc - 1.0f;
            float csum = c[0], cy = c[1], cyy = c[2];
            s0  += csum;
            sx  = fmaf(csum, x, sx);
            sxx = fmaf(csum * x, x, sxx);
            sy  += cy;
            sxy = fmaf(cy, x, sxy);
            syy += cyy;
        }

        s0  = waveReduceAdd(s0);
        sx  = waveReduceAdd(sx);
        sxx = waveReduceAdd(sxx);
        sy  = waveReduceAdd(sy);
        sxy = waveReduceAdd(sxy);
        syy = waveReduceAdd(syy);
        s2  = waveReduceAdd(s2);

        if (lane == 0) {
            // normalize: hm = h / max(sum, 1e-6)
            float T   = s0;
            float Tc  = fmaxf(T, 1e-6f);
            float inv = 1.0f / Tc;
            float s0n = T * inv;
            float sxn = sx * inv, syn = sy * inv;
            float sxxn = sxx * inv, sxyn = sxy * inv, syyn = syy * inv;
            float s2n  = s2 * inv * inv;

            float mx = sxn, my = syn;
            // unbiased weighted covariance numerators
            float cxx = sxxn - 2.0f * mx * sxn + mx * mx * s0n;
            float cxy = sxyn - mx * syn - my * sxn + mx * my * s0n;
            float cyy = syyn - 2.0f * my * syn + my * my * s0n;

            float V1 = s0n + EPSL;
            float V2 = s2n + EPSL;
            float denom = V1 - V2 / V1;
            float id = 1.0f / denom;
            cxx *= id; cxy *= id; cyy *= id;

            // analytic symmetric 2x2 eig, ascending (eigh convention)
            float htr  = 0.5f * (cxx + cyy);
            float hdf  = 0.5f * (cxx - cyy);
            float disc = sqrtf(hdf * hdf + cxy * cxy);
            float l0 = htr - disc;
            float l1 = htr + disc;

            // eigenvector columns V[:,0] (for l0), V[:,1] (for l1)
            float v00, v10, v01, v11;
            if (fabsf(cxy) > 1e-20f) {
                float u0x = cxy, u0y = l0 - cxx;
                float r0 = rsqrtf(u0x * u0x + u0y * u0y);
                u0x *= r0; u0y *= r0;
                float u1x = cxy, u1y = l1 - cxx;
                float r1 = rsqrtf(u1x * u1x + u1y * u1y);
                u1x *= r1; u1y *= r1;
                v00 = u0x; v10 = u0y; v01 = u1x; v11 = u1y;
            } else if (cxx <= cyy) {
                v00 = 1.f; v10 = 0.f; v01 = 0.f; v11 = 1.f;
            } else {
                v00 = 0.f; v10 = 1.f; v01 = 1.f; v11 = 0.f;
            }

            float px = gt[2 * p + 0] - mx;
            float py = gt[2 * p + 1] - my;
            // rotate[j] = sum_i pts[i] * V[j][i]   (einsum 'bni,bnji->bnj')
            float r0v = px * v00 + py * v01;
            float r1v = px * v10 + py * v11;
            float sc0 = r0v * rsqrtf(l0 + EPSL);
            float sc1 = r1v * rsqrtf(l1 + EPSL);

            tcontrib = sc0 * sc0 + sc1 * sc1;   // -> mean over [b,n,2]
            econtrib = fabsf(l0) + fabsf(l1);   // -> mean over [b,n]
        }
    }

    if (lane == 0) { sT[wave] = tcontrib; sE[wave] = econtrib; }
    __syncthreads();
    if (threadIdx.x == 0) {
        float a = 0.f, b = 0.f;
#pragma unroll
        for (int i = 0; i < 8; ++i) { a += sT[i]; b += sE[i]; }
        ws[2 * blockIdx.x + 0] = a;
        ws[2 * blockIdx.x + 1] = b;
    }
}

// Stage 2: deterministic final reduction over per-block partials.
__global__ __launch_bounds__(256) void star_stage2(
    const float* __restrict__ ws, float* __restrict__ out,
    int nblocks, int n_pairs)
{
    __shared__ float sA[256];
    __shared__ float sB[256];
    float a = 0.f, b = 0.f;
    for (int i = threadIdx.x; i < nblocks; i += 256) {
        a += ws[2 * i + 0];
        b += ws[2 * i + 1];
    }
    sA[threadIdx.x] = a; sB[threadIdx.x] = b;
    __syncthreads();
    for (int s = 128; s > 0; s >>= 1) {
        if ((int)threadIdx.x < s) {
            sA[threadIdx.x] += sA[threadIdx.x + s];
            sB[threadIdx.x] += sB[threadIdx.x + s];
        }
        __syncthreads();
    }
    if (threadIdx.x == 0) {
        float inv_pairs = 1.0f / (float)n_pairs;
        // loss = mean(scale^2 over [b,n,2]) + mean(|l0|+|l1| over [b,n])
        out[0] = sA[0] * 0.5f * inv_pairs + sB[0] * inv_pairs;
    }
}

extern "C" void kernel_launch(void* const* d_in, const int* in_sizes, int n_in,
                              void* d_out, int out_size, void* d_ws, size_t ws_size,
                              hipStream_t stream) {
    const float* hm = (const float*)d_in[0];   // [128, 98, 64, 64] fp32
    const float* gt = (const float*)d_in[1];   // [128, 98, 2] fp32
    (void)n_in; (void)out_size; (void)ws_size;

    const int n_pairs = in_sizes[1] / 2;           // 12544
    const int nblocks = (n_pairs + 7) / 8;         // 8 waves (pairs) per block

    float* ws = (float*)d_ws;
    star_stage1<<<nblocks, 256, 0, stream>>>(hm, gt, ws, n_pairs);
    star_stage2<<<1, 256, 0, stream>>>(ws, (float*)d_out, nblocks, n_pairs);
}